// EncoderMemNN_5059471474936
// MI455X (gfx1250) — compile-verified
//
#include <hip/hip_runtime.h>
#include <math.h>

// ---------------------------------------------------------------------------
// EncoderMemNN forward for MI455X (gfx1250, wave32, WMMA).
//   m   = sum_l wordemb[entries] * (entries != 0)          [50000,256]
//   u   = query @ Qgate_w^T + b                            [64,256]
//   3x: logits = u @ m^T / 16 ; prob = softmax(logits) ; u = prob @ m
// fp32 throughout (matches reference numerics); matrix ops via
// V_WMMA_F32_16X16X4_F32. Memory-bound: m (51.2MB) stays resident in the
// 192MB L2 across all 6 GEMM passes.
// ---------------------------------------------------------------------------

typedef float v2f __attribute__((ext_vector_type(2)));
typedef float v4f __attribute__((ext_vector_type(4)));
typedef float v8f __attribute__((ext_vector_type(8)));

#define M_ENT     50000
#define L_TOK     8
#define EMB       256
#define BATCH     64
#define NHID      1024
#define HOPS      3
#define INV_SCALE 0.0625f          // 1/sqrt(256)
#define SPLITK    128
#define CHUNK     392              // even -> 8B-aligned v2f loads; 128*392 >= 50000
#define NTILES    (M_ENT / 16)     // 3125

__device__ __forceinline__ v8f wmma4(v2f a, v2f b, v8f c) {
  // (neg_a, A, neg_b, B, c_mod, C, reuse_a, reuse_b)
  return __builtin_amdgcn_wmma_f32_16x16x4_f32(false, a, false, b,
                                               (short)0, c, false, false);
}

// ---------------------------------------------------------------------------
// m[i,:] = sum over 8 tokens (tok != PAD) of wordemb[tok,:]
// one wave per memory entry; each lane owns 8 consecutive floats.
// ---------------------------------------------------------------------------
__global__ __launch_bounds__(256)
void k_build_m(const int* __restrict__ entries,
               const float* __restrict__ wordemb,
               float* __restrict__ m) {
  const int wave = threadIdx.x >> 5;
  const int lane = threadIdx.x & 31;
  const int row  = blockIdx.x * 8 + wave;
  if (row >= M_ENT) return;
  const int eb = lane * 8;
  v4f acc0 = {0.f, 0.f, 0.f, 0.f};
  v4f acc1 = {0.f, 0.f, 0.f, 0.f};
#pragma unroll
  for (int l = 0; l < L_TOK; ++l) {
    const int tok = entries[row * L_TOK + l];   // uniform across the wave
    if (tok != 0) {
      const float* w = wordemb + (size_t)tok * EMB + eb;
      acc0 += *(const v4f*)(w);
      acc1 += *(const v4f*)(w + 4);
    }
  }
  float* out = m + (size_t)row * EMB + eb;
  *(v4f*)(out)     = acc0;
  *(v4f*)(out + 4) = acc1;
}

// ---------------------------------------------------------------------------
// u[64,256] = query[64,1024] @ Qgate_w[256,1024]^T + Qgate_b
// single block, 8 waves; wave w owns N-tiles {2w, 2w+1}, all 4 M-tiles.
// ---------------------------------------------------------------------------
__global__ __launch_bounds__(256)
void k_qgate(const float* __restrict__ query,
             const float* __restrict__ Wg,
             const float* __restrict__ bg,
             float* __restrict__ u) {
  const int wave = threadIdx.x >> 5;
  const int lane = threadIdx.x & 31;
  const int l16  = lane & 15;
  const int hi   = lane >> 4;      // 0 or 1
  const int kA   = hi * 2;         // K sub-index this lane supplies
  v8f acc[4][2];
#pragma unroll
  for (int mt = 0; mt < 4; ++mt)
#pragma unroll
    for (int t = 0; t < 2; ++t)
      acc[mt][t] = (v8f){0.f,0.f,0.f,0.f,0.f,0.f,0.f,0.f};

  for (int k0 = 0; k0 < NHID; k0 += 4) {
    v2f a[4];
#pragma unroll
    for (int mt = 0; mt < 4; ++mt)
      a[mt] = *(const v2f*)(query + (size_t)(mt * 16 + l16) * NHID + k0 + kA);
#pragma unroll
    for (int t = 0; t < 2; ++t) {
      const int e = (wave * 2 + t) * 16 + l16;
      v2f b = *(const v2f*)(Wg + (size_t)e * NHID + k0 + kA);
#pragma unroll
      for (int mt = 0; mt < 4; ++mt)
        acc[mt][t] = wmma4(a[mt], b, acc[mt][t]);
    }
  }
#pragma unroll
  for (int t = 0; t < 2; ++t) {
    const int e = (wave * 2 + t) * 16 + l16;
    const float bias = bg[e];
#pragma unroll
    for (int mt = 0; mt < 4; ++mt)
#pragma unroll
      for (int r = 0; r < 8; ++r) {
        const int row = mt * 16 + r + hi * 8;   // C/D layout
        u[(size_t)row * EMB + e] = acc[mt][t][r] + bias;
      }
  }
}

// ---------------------------------------------------------------------------
// logits[64,50000] = u[64,256] @ m[50000,256]^T * (1/16)
// Per block: stage u in LDS pre-packed in WMMA A-fragment order (64KB,
// conflict-free ds_load_b64), then each wave computes one 16-wide N-tile
// against all 4 M-tiles over K=256.
// ---------------------------------------------------------------------------
__global__ __launch_bounds__(256)
void k_logits(const float* __restrict__ u,
              const float* __restrict__ m,
              float* __restrict__ logits) {
  __shared__ float u_lds[BATCH * EMB];   // 65536 B, packed as [kv][mt][lane]{2}
  const int wave = threadIdx.x >> 5;
  const int lane = threadIdx.x & 31;
  const int l16  = lane & 15;
  const int hi   = lane >> 4;
  const int kA   = hi * 2;

  // pack u into A-fragment order: pair p = ((kv*4 + mt)*32 + ln)
  for (int p = threadIdx.x; p < (BATCH * EMB) / 2; p += 256) {
    const int kv  = p >> 7;
    const int mt  = (p >> 5) & 3;
    const int ln  = p & 31;
    const int row = mt * 16 + (ln & 15);
    const int col = kv * 4 + ((ln >> 4) << 1);
    v2f v = *(const v2f*)(u + (size_t)row * EMB + col);
    u_lds[2 * p]     = v.x;
    u_lds[2 * p + 1] = v.y;
  }
  __syncthreads();

  const int ntile = blockIdx.x * 8 + wave;
  if (ntile >= NTILES) return;
  const int n = ntile * 16 + l16;          // memory-entry column this lane owns

  v8f acc[4];
#pragma unroll
  for (int mt = 0; mt < 4; ++mt)
    acc[mt] = (v8f){0.f,0.f,0.f,0.f,0.f,0.f,0.f,0.f};

  const float* mrow = m + (size_t)n * EMB;
  for (int k0 = 0; k0 < EMB; k0 += 4) {
    const int kv = k0 >> 2;
    v2f b = *(const v2f*)(mrow + k0 + kA);   // B fragment: m^T tile
#pragma unroll
    for (int mt = 0; mt < 4; ++mt) {
      v2f a = *(const v2f*)&u_lds[(((kv * 4 + mt) * 32) + lane) * 2];
      acc[mt] = wmma4(a, b, acc[mt]);
    }
  }
#pragma unroll
  for (int mt = 0; mt < 4; ++mt)
#pragma unroll
    for (int r = 0; r < 8; ++r) {
      const int row = mt * 16 + r + hi * 8;
      logits[(size_t)row * M_ENT + n] = acc[mt][r] * INV_SCALE;
    }
}

// ---------------------------------------------------------------------------
// row softmax over 50000; one block per batch row.
// ---------------------------------------------------------------------------
__global__ __launch_bounds__(256)
void k_softmax(const float* __restrict__ logits, float* __restrict__ prob) {
  __shared__ float red[256];
  __shared__ float s_max, s_sum;
  const int row = blockIdx.x;
  const int tid = threadIdx.x;
  const float* x = logits + (size_t)row * M_ENT;
  float* p = prob + (size_t)row * M_ENT;

  float mx = -3.402823466e38f;
  for (int j = tid; j < M_ENT; j += 256) mx = fmaxf(mx, x[j]);
  red[tid] = mx;
  __syncthreads();
  for (int s = 128; s > 0; s >>= 1) {
    if (tid < s) red[tid] = fmaxf(red[tid], red[tid + s]);
    __syncthreads();
  }
  if (tid == 0) s_max = red[0];
  __syncthreads();
  mx = s_max;

  float sum = 0.f;
  for (int j = tid; j < M_ENT; j += 256) {
    const float e = __expf(x[j] - mx);
    p[j] = e;
    sum += e;
  }
  red[tid] = sum;
  __syncthreads();
  for (int s = 128; s > 0; s >>= 1) {
    if (tid < s) red[tid] += red[tid + s];
    __syncthreads();
  }
  if (tid == 0) s_sum = red[0];
  __syncthreads();
  const float inv = 1.f / s_sum;
  for (int j = tid; j < M_ENT; j += 256) p[j] *= inv;
}

// ---------------------------------------------------------------------------
// split-K partials: partial[s] = prob[:, chunk_s] @ m[chunk_s, :]
// block s handles rows [s*CHUNK, min(...,50000)); 8 waves x (4 M x 2 N) tiles.
// ---------------------------------------------------------------------------
__global__ __launch_bounds__(256)
void k_av_partial(const float* __restrict__ prob,
                  const float* __restrict__ m,
                  float* __restrict__ partial) {
  const int s     = blockIdx.x;
  const int j_beg = s * CHUNK;
  const int j_end = (j_beg + CHUNK < M_ENT) ? (j_beg + CHUNK) : M_ENT;
  const int wave  = threadIdx.x >> 5;
  const int lane  = threadIdx.x & 31;
  const int l16   = lane & 15;
  const int hi    = lane >> 4;
  const int kA    = hi * 2;

  v8f acc[4][2];
#pragma unroll
  for (int mt = 0; mt < 4; ++mt)
#pragma unroll
    for (int t = 0; t < 2; ++t)
      acc[mt][t] = (v8f){0.f,0.f,0.f,0.f,0.f,0.f,0.f,0.f};

  for (int j0 = j_beg; j0 < j_end; j0 += 4) {
    const int ja = j0 + kA;
    const bool v0 = (ja     < j_end);
    const bool v1 = (ja + 1 < j_end);
    v2f a[4];
#pragma unroll
    for (int mt = 0; mt < 4; ++mt) {
      const float* pr = prob + (size_t)(mt * 16 + l16) * M_ENT;
      v2f av;
      av.x = v0 ? pr[ja]     : 0.f;
      av.y = v1 ? pr[ja + 1] : 0.f;
      a[mt] = av;
    }
#pragma unroll
    for (int t = 0; t < 2; ++t) {
      const int e = (wave * 2 + t) * 16 + l16;
      v2f b;
      b.x = v0 ? m[(size_t)ja * EMB + e]       : 0.f;
      b.y = v1 ? m[(size_t)(ja + 1) * EMB + e] : 0.f;
#pragma unroll
      for (int mt = 0; mt < 4; ++mt)
        acc[mt][t] = wmma4(a[mt], b, acc[mt][t]);
    }
  }

  float* out = partial + (size_t)s * (BATCH * EMB);
#pragma unroll
  for (int t = 0; t < 2; ++t) {
    const int e = (wave * 2 + t) * 16 + l16;
#pragma unroll
    for (int mt = 0; mt < 4; ++mt)
#pragma unroll
      for (int r = 0; r < 8; ++r) {
        const int row = mt * 16 + r + hi * 8;
        out[(size_t)row * EMB + e] = acc[mt][t][r];
      }
  }
}

// deterministic fixed-order reduction of split-K partials
__global__ __launch_bounds__(256)
void k_reduce_u(const float* __restrict__ partial, float* __restrict__ u) {
  const int idx = blockIdx.x * 256 + threadIdx.x;
  if (idx >= BATCH * EMB) return;
  float acc = 0.f;
  for (int p = 0; p < SPLITK; ++p)
    acc += partial[(size_t)p * (BATCH * EMB) + idx];
  u[idx] = acc;
}

// ---------------------------------------------------------------------------
extern "C" void kernel_launch(void* const* d_in, const int* in_sizes, int n_in,
                              void* d_out, int out_size, void* d_ws, size_t ws_size,
                              hipStream_t stream) {
  (void)in_sizes; (void)n_in; (void)out_size; (void)ws_size;

  const int*   entries = (const int*)  d_in[0];
  const float* query   = (const float*)d_in[1];
  const float* wordemb = (const float*)d_in[2];
  const float* Qgate_w = (const float*)d_in[3];
  const float* Qgate_b = (const float*)d_in[4];

  float* out_u      = (float*)d_out;            // [64,256]
  float* out_logits = out_u + BATCH * EMB;      // [64,50000] (last hop)

  // workspace layout (floats): m | prob | partials | u ping | u pong
  float* ws    = (float*)d_ws;
  float* m_    = ws;                                      // 12.80M
  float* prob  = m_   + (size_t)M_ENT * EMB;              //  3.20M
  float* part  = prob + (size_t)BATCH * M_ENT;            //  2.10M
  float* u_a   = part + (size_t)SPLITK * BATCH * EMB;
  float* u_b   = u_a  + BATCH * EMB;

  k_build_m<<<(M_ENT + 7) / 8, 256, 0, stream>>>(entries, wordemb, m_);
  k_qgate<<<1, 256, 0, stream>>>(query, Qgate_w, Qgate_b, u_a);

  float* u_cur = u_a;
  float* u_nxt = u_b;
  for (int hop = 0; hop < HOPS; ++hop) {
    k_logits<<<(NTILES + 7) / 8, 256, 0, stream>>>(u_cur, m_, out_logits);
    k_softmax<<<BATCH, 256, 0, stream>>>(out_logits, prob);
    k_av_partial<<<SPLITK, 256, 0, stream>>>(prob, m_, part);
    float* dst = (hop == HOPS - 1) ? out_u : u_nxt;
    k_reduce_u<<<(BATCH * EMB + 255) / 256, 256, 0, stream>>>(part, dst);
    float* tmp = u_cur; u_cur = u_nxt; u_nxt = tmp;
  }
}